// MentionScore_42451456753704
// MI455X (gfx1250) — compile-verified
//
#include <hip/hip_runtime.h>
#include <hip/hip_bf16.h>

// ---------------- problem constants (match reference) ----------------
#define BB   16
#define TT   2048
#define EE   1024
#define SS   2000
#define HID  150
#define GI   (3 * EE)

// ---------------- MLP kernel tiling ----------------
#define BQ    32            // rows (M) per workgroup: 2 M-tiles of 16
#define NW    10            // waves per workgroup -> 10 N-tiles of 16 = 160 >= HID
#define TPB   (NW * 32)     // 320 threads
#define CHUNK 512           // K staging chunk (divides 1024 and 3072)
#define LDSK  (CHUNK + 4)   // pad: stride 516 -> bank stride 4, conflict free
#define HPAD  164           // h1/h2 row stride (164 mod 64 = 36, conflict free)
#define K2P   152           // layer-2 K padded to multiple of 4 (>= 150)

// ---------------- scan tiling ----------------
#define CT    256           // tokens per scan chunk
#define NCH   (TT / CT)     // 8 chunks

typedef float v2f __attribute__((ext_vector_type(2)));
typedef float v8f __attribute__((ext_vector_type(8)));

__device__ __forceinline__ v8f wmma_f32(v2f a, v2f b, v8f c) {
  // V_WMMA_F32_16X16X4_F32:  D = A(16x4) x B(4x16) + C(16x16), all fp32
  return __builtin_amdgcn_wmma_f32_16x16x4_f32(
      /*neg_a=*/false, a, /*neg_b=*/false, b,
      /*c_mod=*/(short)0, c, /*reuse_a=*/false, /*reuse_b=*/false);
}

// =====================================================================
// 3-layer MLP: out[row] = W3 . relu( relu(X[row,:]@W1 + b1) @ W2 + b2 ) + b3
// One workgroup = 32 rows; 10 waves each own one 16-wide N tile and
// two 16-row M tiles (B fragment reused by 2 WMMAs -> higher density).
// =====================================================================
__global__ __launch_bounds__(TPB)
void mlp3_wmma_kernel(const float* __restrict__ X, int kdim,
                      const float* __restrict__ W1, const float* __restrict__ b1,
                      const float* __restrict__ W2, const float* __restrict__ b2,
                      const float* __restrict__ W3, const float* __restrict__ b3,
                      float* __restrict__ out) {
  __shared__ float xs[BQ][LDSK];
  __shared__ float h1[BQ][HPAD];
  __shared__ float h2[BQ][HPAD];

  const int tid  = threadIdx.x;
  const int wave = tid >> 5;
  const int lane = tid & 31;
  const int m    = lane & 15;           // A-matrix row this lane feeds (tile 0)
  const int koff = (lane >> 4) << 1;    // 0 (lanes 0-15) or 2 (lanes 16-31)
  const int n    = wave * 16 + m;       // B-matrix column this lane feeds
  const int nc   = (n < HID) ? n : (HID - 1);   // clamp for pad columns
  const long row0 = (long)blockIdx.x * BQ;

  // zero h1/h2 (pad region must stay zero so padded-K WMMA adds nothing)
  for (int i = tid; i < BQ * HPAD; i += TPB) {
    (&h1[0][0])[i] = 0.0f;
    (&h2[0][0])[i] = 0.0f;
  }

  // ----------------- layer 1: [32 x kdim] @ [kdim x HID] -----------------
  v8f acc0 = {0.f, 0.f, 0.f, 0.f, 0.f, 0.f, 0.f, 0.f};
  v8f acc1 = {0.f, 0.f, 0.f, 0.f, 0.f, 0.f, 0.f, 0.f};
  for (int c0 = 0; c0 < kdim; c0 += CHUNK) {
    __syncthreads();
    for (int i = tid; i < BQ * CHUNK; i += TPB) {
      const int r  = i >> 9;            // / CHUNK
      const int cc = i & (CHUNK - 1);
      xs[r][cc] = X[(row0 + r) * (long)kdim + c0 + cc];
    }
    __syncthreads();
#pragma unroll 4
    for (int k = 0; k < CHUNK; k += 4) {
      v2f a0, a1, b;
      a0.x = xs[m][k + koff];
      a0.y = xs[m][k + koff + 1];
      a1.x = xs[m + 16][k + koff];
      a1.y = xs[m + 16][k + koff + 1];
      const float* wp = W1 + (long)(c0 + k + koff) * HID + nc;
      b.x = wp[0];
      b.y = wp[HID];
      acc0 = wmma_f32(a0, b, acc0);
      acc1 = wmma_f32(a1, b, acc1);
    }
  }
  {   // scatter both C tiles to LDS with bias + relu (16x16 f32 D layout)
    const int mrow = (lane >> 4) * 8;
    const int ncol = wave * 16 + (lane & 15);
    if (ncol < HID) {
      const float bias = b1[ncol];
#pragma unroll
      for (int i = 0; i < 8; ++i) {
        const float v0 = acc0[i] + bias;
        const float v1 = acc1[i] + bias;
        h1[mrow + i][ncol]      = (v0 > 0.0f) ? v0 : 0.0f;
        h1[16 + mrow + i][ncol] = (v1 > 0.0f) ? v1 : 0.0f;
      }
    }
  }
  __syncthreads();

  // ----------------- layer 2: [32 x 150] @ [150 x HID] -----------------
  v8f acc2 = {0.f, 0.f, 0.f, 0.f, 0.f, 0.f, 0.f, 0.f};
  v8f acc3 = {0.f, 0.f, 0.f, 0.f, 0.f, 0.f, 0.f, 0.f};
#pragma unroll 4
  for (int k = 0; k < K2P; k += 4) {
    v2f a0, a1, b;
    a0.x = h1[m][k + koff];             // zero in pad region
    a0.y = h1[m][k + koff + 1];
    a1.x = h1[m + 16][k + koff];
    a1.y = h1[m + 16][k + koff + 1];
    const int g0 = k + koff;
    const int g0c = (g0 < HID) ? g0 : (HID - 1);       // clamp (a==0 there)
    const int g1c = (g0 + 1 < HID) ? g0 + 1 : (HID - 1);
    b.x = W2[g0c * HID + nc];
    b.y = W2[g1c * HID + nc];
    acc2 = wmma_f32(a0, b, acc2);
    acc3 = wmma_f32(a1, b, acc3);
  }
  {
    const int mrow = (lane >> 4) * 8;
    const int ncol = wave * 16 + (lane & 15);
    if (ncol < HID) {
      const float bias = b2[ncol];
#pragma unroll
      for (int i = 0; i < 8; ++i) {
        const float v0 = acc2[i] + bias;
        const float v1 = acc3[i] + bias;
        h2[mrow + i][ncol]      = (v0 > 0.0f) ? v0 : 0.0f;
        h2[16 + mrow + i][ncol] = (v1 > 0.0f) ? v1 : 0.0f;
      }
    }
  }
  __syncthreads();

  // ----------------- layer 3: [32 x 150] . W3[150] -----------------
  if (tid < BQ) {
    float s = 0.0f;
#pragma unroll 5
    for (int j = 0; j < HID; ++j) s += h2[tid][j] * W3[j];
    out[row0 + tid] = s + b3[0];
  }
}

// =====================================================================
// Pass 1 of scan: per-chunk partial sums of weighted = X * attn
// grid = B * NCH, 256 threads, float4 per thread over E
// =====================================================================
__global__ __launch_bounds__(256)
void chunk_sums_kernel(const float* __restrict__ X, const float* __restrict__ attn,
                       float* __restrict__ chunkSum) {
  const int b  = blockIdx.x / NCH;
  const int ch = blockIdx.x % NCH;
  const int e4 = threadIdx.x;           // E/4 == 256
  const float4* xp = (const float4*)(X + ((long)b * TT + (long)ch * CT) * EE);
  const float*  ap = attn + (long)b * TT + (long)ch * CT;
  float4 s = make_float4(0.f, 0.f, 0.f, 0.f);
  for (int t = 0; t < CT; ++t) {
    const float4 v = xp[(long)t * (EE / 4) + e4];
    const float  a = ap[t];
    s.x += v.x * a; s.y += v.y * a; s.z += v.z * a; s.w += v.w * a;
  }
  ((float4*)chunkSum)[((long)b * NCH + ch) * (EE / 4) + e4] = s;
}

// =====================================================================
// Pass 2 of scan: sequential scan within each chunk + chunk offset,
// writes csum[b][t+1][e]; chunk 0 also writes csum[b][0][e] = 0.
// =====================================================================
__global__ __launch_bounds__(256)
void scan_chunks_kernel(const float* __restrict__ X, const float* __restrict__ attn,
                        const float* __restrict__ chunkSum, float* __restrict__ csum) {
  const int b  = blockIdx.x / NCH;
  const int ch = blockIdx.x % NCH;
  const int e4 = threadIdx.x;
  const float4* cs = (const float4*)chunkSum;
  float4 run = make_float4(0.f, 0.f, 0.f, 0.f);
  for (int c = 0; c < ch; ++c) {
    const float4 v = cs[((long)b * NCH + c) * (EE / 4) + e4];
    run.x += v.x; run.y += v.y; run.z += v.z; run.w += v.w;
  }
  float4* cp = (float4*)(csum + (long)b * (TT + 1) * EE);
  if (ch == 0) cp[e4] = make_float4(0.f, 0.f, 0.f, 0.f);   // csum row 0
  const float4* xp = (const float4*)(X + ((long)b * TT + (long)ch * CT) * EE);
  const float*  ap = attn + (long)b * TT + (long)ch * CT;
  for (int t = 0; t < CT; ++t) {
    const float4 v = xp[(long)t * (EE / 4) + e4];
    const float  a = ap[t];
    run.x += v.x * a; run.y += v.y * a; run.z += v.z * a; run.w += v.w * a;
    cp[((long)(ch * CT + t) + 1) * (EE / 4) + e4] = run;
  }
}

// =====================================================================
// Span gather: span_emb[b,s,:] = [x[start], x[end], csum[end+1]-csum[start]]
// zeros for spans >= num_spans[b]. grid = B*S blocks, 256 threads (float4).
// =====================================================================
__global__ __launch_bounds__(256)
void gather_spans_kernel(const float* __restrict__ X, const float* __restrict__ csum,
                         const int* __restrict__ starts, const int* __restrict__ lengths,
                         const int* __restrict__ nspans, float* __restrict__ span_emb) {
  const long bs = blockIdx.x;
  const int b = (int)(bs / SS);
  const int s = (int)(bs % SS);
  const int e4 = threadIdx.x;                  // E/4 == 256
  float4* op = (float4*)(span_emb + bs * (long)GI);
  if (s >= nspans[b]) {
    const float4 z = make_float4(0.f, 0.f, 0.f, 0.f);
    op[e4] = z;
    op[(EE / 4) + e4] = z;
    op[2 * (EE / 4) + e4] = z;
    return;
  }
  const int st = starts[b * SS + s];
  const int en = st + lengths[b * SS + s];     // inclusive end, < T
  const float4* xb = (const float4*)(X + (long)b * TT * EE);
  const float4* cb = (const float4*)(csum + (long)b * (TT + 1) * EE);
  op[e4]              = xb[(long)st * (EE / 4) + e4];
  op[(EE / 4) + e4]   = xb[(long)en * (EE / 4) + e4];
  const float4 hi = cb[(long)(en + 1) * (EE / 4) + e4];
  const float4 lo = cb[(long)st * (EE / 4) + e4];
  op[2 * (EE / 4) + e4] = make_float4(hi.x - lo.x, hi.y - lo.y, hi.z - lo.z, hi.w - lo.w);
}

// =====================================================================
extern "C" void kernel_launch(void* const* d_in, const int* in_sizes, int n_in,
                              void* d_out, int out_size, void* d_ws, size_t ws_size,
                              hipStream_t stream) {
  const float* X       = (const float*)d_in[0];   // [B,T,E]
  const int*   starts  = (const int*)d_in[1];     // [B,S]
  const int*   lengths = (const int*)d_in[2];     // [B,S]
  const int*   nspans  = (const int*)d_in[3];     // [B]
  const float* Wa1 = (const float*)d_in[4];
  const float* ba1 = (const float*)d_in[5];
  const float* Wa2 = (const float*)d_in[6];
  const float* ba2 = (const float*)d_in[7];
  const float* Wa3 = (const float*)d_in[8];
  const float* ba3 = (const float*)d_in[9];
  const float* Ws1 = (const float*)d_in[10];
  const float* bs1 = (const float*)d_in[11];
  const float* Ws2 = (const float*)d_in[12];
  const float* bs2 = (const float*)d_in[13];
  const float* Ws3 = (const float*)d_in[14];
  const float* bs3 = (const float*)d_in[15];

  // workspace layout (floats): attn[B*T] | chunkSum[B*NCH*E] | csum[B*(T+1)*E]
  float* ws       = (float*)d_ws;
  float* attn     = ws;
  float* chunkSum = attn + (long)BB * TT;
  float* csum     = chunkSum + (long)BB * NCH * EE;

  float* span_emb = (float*)d_out;                        // [B,S,3E]
  float* scores   = span_emb + (long)BB * SS * GI;        // [B,S,1]

  // 1) token attention MLP -> attn[B*T]
  mlp3_wmma_kernel<<<(BB * TT) / BQ, TPB, 0, stream>>>(
      X, EE, Wa1, ba1, Wa2, ba2, Wa3, ba3, attn);

  // 2) weighted prefix sums (two-level scan) -> csum[B,T+1,E]
  chunk_sums_kernel<<<BB * NCH, 256, 0, stream>>>(X, attn, chunkSum);
  scan_chunks_kernel<<<BB * NCH, 256, 0, stream>>>(X, attn, chunkSum, csum);

  // 3) span gather + concat + mask -> span_emb (d_out)
  gather_spans_kernel<<<BB * SS, 256, 0, stream>>>(
      X, csum, starts, lengths, nspans, span_emb);

  // 4) mention scoring MLP on span_emb -> scores (d_out tail)
  mlp3_wmma_kernel<<<(BB * SS) / BQ, TPB, 0, stream>>>(
      span_emb, GI, Ws1, bs1, Ws2, bs2, Ws3, bs3, scores);
}